// FourierMMLayer_22454089024219
// MI455X (gfx1250) — compile-verified
//
#include <hip/hip_runtime.h>

// FNet-style Fourier mixing: out = Re(F_S * x * F_H), x real [32,512,768].
// Decomposed as out[b] = ReS @ (x[b] @ ReH)  +  (-ImS) @ (x[b] @ ImH)
// Fused per (batch, 32-column block): stage-1 result lives in LDS (128 KB),
// both stages on V_WMMA_F32_16X16X4_F32 (f32 matrix cores, wave32).
// This revision keeps all 4(M) x 2(N) tile accumulators live through a single
// K loop so A/B fragments are loaded once per K-step (0.75 loads per wmma).

#define SEQ 512
#define HID 768
#define KB  32                  // output-column block held in LDS
#define NKB (HID / KB)          // 24 column blocks
#define NB  32                  // batch

typedef __attribute__((ext_vector_type(2))) float v2f;
typedef __attribute__((ext_vector_type(8))) float v8f;

__device__ __forceinline__ v8f wmma4_f32(v2f a, v2f b, v8f c) {
    // 8 args: (neg_a, A, neg_b, B, c_mod, C, reuse_a, reuse_b)
    return __builtin_amdgcn_wmma_f32_16x16x4_f32(
        false, a, false, b, (short)0, c, false, false);
}

// Split interleaved complex128 -> f32 re/im planes (im optionally negated).
__global__ void split_complex_kernel(const double* __restrict__ z,
                                     float* __restrict__ re,
                                     float* __restrict__ im,
                                     int n, float im_scale) {
    int i = blockIdx.x * blockDim.x + threadIdx.x;
    if (i < n) {
        re[i] = (float)z[2 * i];
        im[i] = im_scale * (float)z[2 * i + 1];
    }
}

__global__ __launch_bounds__(256)
void fourier_fused_kernel(const float* __restrict__ x,
                          const float* __restrict__ ReS,
                          const float* __restrict__ nImS,   // pre-negated Im(F_S)
                          const float* __restrict__ ReH,
                          const float* __restrict__ ImH,
                          float* __restrict__ out) {
    extern __shared__ float smem[];
    float* Tr = smem;                 // [SEQ][KB]
    float* Ti = smem + SEQ * KB;      // [SEQ][KB]

    const int kb   = blockIdx.x;          // column block 0..NKB-1
    const int b    = blockIdx.y;          // batch
    const int lane = threadIdx.x & 31;
    const int wave = threadIdx.x >> 5;    // 0..7

    const int ln   = lane & 15;           // N (or M) index within a 16-tile
    const int lk2  = (lane >> 4) * 2;     // K sub-offset per wmma layout
    const int vrow = 8 * (lane >> 4);     // C/D: M offset for this lane half

    const float* xb    = x + (size_t)b * SEQ * HID;
    const int    nc0   = kb * KB;
    const int    mbase = wave * 64;       // 64 of the 512 rows per wave

    // ---- Phase 1: Tr/Ti = x[b] (512x768) @ {ReH,ImH}[:, nc0:nc0+KB] -> LDS
    {
        v8f accr[4][2] = {{{}, {}}, {{}, {}}, {{}, {}}, {{}, {}}};
        v8f acci[4][2] = {{{}, {}}, {{}, {}}, {{}, {}}, {{}, {}}};
        for (int k0 = 0; k0 < HID; k0 += 4) {
            const int ka = k0 + lk2;
            v2f a[4];
#pragma unroll
            for (int mt = 0; mt < 4; ++mt)
                a[mt] = *(const v2f*)(xb + (size_t)(mbase + mt * 16 + ln) * HID + ka);
            v2f br[2], bi[2];
#pragma unroll
            for (int nt = 0; nt < 2; ++nt) {
                const int n = nc0 + nt * 16 + ln;
                br[nt].x = ReH[(size_t)ka * HID + n];
                br[nt].y = ReH[(size_t)(ka + 1) * HID + n];
                bi[nt].x = ImH[(size_t)ka * HID + n];
                bi[nt].y = ImH[(size_t)(ka + 1) * HID + n];
            }
#pragma unroll
            for (int mt = 0; mt < 4; ++mt)
#pragma unroll
                for (int nt = 0; nt < 2; ++nt) {
                    accr[mt][nt] = wmma4_f32(a[mt], br[nt], accr[mt][nt]);
                    acci[mt][nt] = wmma4_f32(a[mt], bi[nt], acci[mt][nt]);
                }
        }
#pragma unroll
        for (int mt = 0; mt < 4; ++mt)
#pragma unroll
            for (int nt = 0; nt < 2; ++nt) {
                const int nl = nt * 16 + ln;
#pragma unroll
                for (int v = 0; v < 8; ++v) {
                    const int m = mbase + mt * 16 + v + vrow;
                    Tr[m * KB + nl] = accr[mt][nt][v];
                    Ti[m * KB + nl] = acci[mt][nt][v];
                }
            }
    }
    __syncthreads();

    // ---- Phase 2: out[b][:, nc0:nc0+KB] = ReS @ Tr + (-ImS) @ Ti
    {
        v8f acc[4][2] = {{{}, {}}, {{}, {}}, {{}, {}}, {{}, {}}};
        for (int k0 = 0; k0 < SEQ; k0 += 4) {
            const int ka = k0 + lk2;
            v2f ar[4], ai[4];
#pragma unroll
            for (int mt = 0; mt < 4; ++mt) {
                const size_t row = (size_t)(mbase + mt * 16 + ln) * SEQ + ka;
                ar[mt] = *(const v2f*)(ReS + row);
                ai[mt] = *(const v2f*)(nImS + row);
            }
            v2f br[2], bi[2];
#pragma unroll
            for (int nt = 0; nt < 2; ++nt) {
                const int nl = nt * 16 + ln;
                br[nt].x = Tr[ka * KB + nl];
                br[nt].y = Tr[(ka + 1) * KB + nl];
                bi[nt].x = Ti[ka * KB + nl];
                bi[nt].y = Ti[(ka + 1) * KB + nl];
            }
#pragma unroll
            for (int mt = 0; mt < 4; ++mt)
#pragma unroll
                for (int nt = 0; nt < 2; ++nt) {
                    acc[mt][nt] = wmma4_f32(ar[mt], br[nt], acc[mt][nt]);
                    acc[mt][nt] = wmma4_f32(ai[mt], bi[nt], acc[mt][nt]);
                }
        }
#pragma unroll
        for (int mt = 0; mt < 4; ++mt)
#pragma unroll
            for (int nt = 0; nt < 2; ++nt) {
                const int nl = nt * 16 + ln;
#pragma unroll
                for (int v = 0; v < 8; ++v) {
                    const int m = mbase + mt * 16 + v + vrow;
                    out[(size_t)b * SEQ * HID + (size_t)m * HID + nc0 + nl] =
                        acc[mt][nt][v];
                }
            }
    }
}

extern "C" void kernel_launch(void* const* d_in, const int* in_sizes, int n_in,
                              void* d_out, int out_size, void* d_ws, size_t ws_size,
                              hipStream_t stream) {
    const float*  x  = (const float*)d_in[0];
    const double* zS = (const double*)d_in[1];   // [512,512] complex128 (re,im pairs)
    const double* zH = (const double*)d_in[2];   // [768,768] complex128
    float* out = (float*)d_out;

    // Workspace: f32 planes of the DFT matrices (6.8 MB total).
    float* ReS  = (float*)d_ws;                  // SEQ*SEQ
    float* nImS = ReS  + SEQ * SEQ;              // SEQ*SEQ (negated imag)
    float* ReH  = nImS + SEQ * SEQ;              // HID*HID
    float* ImH  = ReH  + HID * HID;              // HID*HID

    {
        const int n = SEQ * SEQ;
        split_complex_kernel<<<(n + 255) / 256, 256, 0, stream>>>(zS, ReS, nImS, n, -1.0f);
    }
    {
        const int n = HID * HID;
        split_complex_kernel<<<(n + 255) / 256, 256, 0, stream>>>(zH, ReH, ImH, n, 1.0f);
    }

    dim3 grid(NKB, NB);                                   // 24 x 32 workgroups
    size_t shmem = (size_t)SEQ * KB * 2 * sizeof(float);  // 128 KB LDS
    fourier_fused_kernel<<<grid, 256, shmem, stream>>>(x, ReS, nImS, ReH, ImH, out);
}